// ImageDiscriminator_12773232738670
// MI455X (gfx1250) — compile-verified
//
#include <hip/hip_runtime.h>
#include <hip/hip_bf16.h>

// Problem constants (match reference)
#define B_   8
#define C_   256
#define K_   19
#define HW_  (256 * 256)

typedef __attribute__((ext_vector_type(2))) float v2f;
typedef __attribute__((ext_vector_type(8))) float v8f;
typedef __attribute__((ext_vector_type(4))) int   v4i;

static __device__ __forceinline__ v8f wmma4(v2f a, v2f b, v8f c) {
  // D(16x16,f32) = A(16x4,f32) x B(4x16,f32) + C
  return __builtin_amdgcn_wmma_f32_16x16x4_f32(
      /*neg_a=*/false, a, /*neg_b=*/false, b,
      /*c_mod=*/(short)0, c, /*reuse_a=*/false, /*reuse_b=*/false);
}

// Async global -> LDS copy (16B per lane), tracked by ASYNCcnt.
static __device__ __forceinline__ void async_g2l_b128(const float* g, float* l) {
  __builtin_amdgcn_global_load_async_to_lds_b128((v4i*)g, (v4i*)l, 0, 0);
}

__global__ void zero_kernel(float* __restrict__ p, int n) {
  for (int i = blockIdx.x * blockDim.x + threadIdx.x; i < n;
       i += gridDim.x * blockDim.x)
    p[i] = 0.0f;
}

// ---------------- Stage 1: feat[b,k,c] = sum_hw cam*x / HW ----------------
// Grid: (64, B). Block: 512 (16 waves). Each WG streams 1024 HW positions.
// Double-buffered LDS tiles filled by GLOBAL_LOAD_ASYNC_TO_LDS_B128; every
// wave issues exactly 3 async ops per slab so s_wait_asynccnt(3) pipelines
// slab s+1's loads behind slab s's WMMAs. x and cam each read once from HBM.
#define T1     16   // hw per LDS tile
#define STR1   20   // padded row stride (floats): rows 16B-aligned, bank-spread
#define SLABS1 64   // 64 WGs/batch * 64 * 16 = 65536 hw

__global__ __launch_bounds__(512) void feat_kernel(
    const float* __restrict__ x, const float* __restrict__ cam,
    float* __restrict__ feat /* [B*32, 256] */) {
  __shared__ float sX[2][C_ * STR1];   // 2 * 20480 B
  __shared__ float sC[2][32 * STR1];   // 2 *  2560 B   (46 KB total)
  const int tid  = threadIdx.x;
  const int lane = tid & 31;
  const int wave = tid >> 5;
  const int b    = blockIdx.y;
  const int m    = lane & 15;          // A row (k) / B col (c) within tile
  const int hoff = (lane >> 4) << 1;   // inner-k offset 0 or 2

  const float* xb = x   + (size_t)b * C_ * HW_;
  const float* cb = cam + (size_t)b * K_ * HW_;

  // cam rows 19..31 stay zero in both buffers (pads K to 32)
  for (int i = tid; i < (32 - K_) * STR1; i += 512) {
    sC[0][K_ * STR1 + i] = 0.0f;
    sC[1][K_ * STR1 + i] = 0.0f;
  }

  // per-thread async-copy coordinates
  const int  xc   = tid >> 1;            // x channel row (2 threads/row)
  const int  xq   = (tid & 1) << 3;      // float offset 0 or 8 in the row
  const int  cidx = wave * 5 + lane;     // cam b128 index, lanes 0..4 only
  const bool con  = (lane < 5) && (cidx < K_ * (T1 / 4));   // 76 pieces
  const int  ck   = cidx >> 2;
  const int  cq   = (cidx & 3) << 2;

  auto issue = [&](int s, int buf) {
    const int hw0 = (blockIdx.x * SLABS1 + s) * T1;
    const float* gx = xb + (size_t)xc * HW_ + hw0 + xq;
    async_g2l_b128(gx,     &sX[buf][xc * STR1 + xq]);
    async_g2l_b128(gx + 4, &sX[buf][xc * STR1 + xq + 4]);
    if (con)   // >=1 active lane in every wave -> uniform 3 ops/wave/slab
      async_g2l_b128(cb + (size_t)ck * HW_ + hw0 + cq, &sC[buf][ck * STR1 + cq]);
  };

  v8f acc0 = {0, 0, 0, 0, 0, 0, 0, 0};   // k rows 0..15
  v8f acc1 = {0, 0, 0, 0, 0, 0, 0, 0};   // k rows 16..31 (19..31 stay 0)

  issue(0, 0);
  for (int s = 0; s < SLABS1; ++s) {
    const int p = s & 1;
    if (s + 1 < SLABS1) {
      issue(s + 1, 1 - p);                   // prefetch next slab
      __builtin_amdgcn_s_wait_asynccnt(3);   // slab s's 3 ops are done
    } else {
      __builtin_amdgcn_s_wait_asynccnt(0);
    }
    __syncthreads();   // all waves' slab-s data visible in LDS
#pragma unroll
    for (int st = 0; st < T1 / 4; ++st) {
      const int h = st * 4 + hoff;
      v2f a0 = *(const v2f*)&sC[p][m * STR1 + h];          // cam rows 0..15
      v2f a1 = *(const v2f*)&sC[p][(16 + m) * STR1 + h];   // cam rows 16..31
      v2f bb = *(const v2f*)&sX[p][(wave * 16 + m) * STR1 + h];
      acc0 = wmma4(a0, bb, acc0);
      acc1 = wmma4(a1, bb, acc1);
    }
    __syncthreads();   // readers done before buffer p is refilled
  }

  // Reduce partial tiles into feat (scaled by 1/HW)
  const float sc = 1.0f / (float)HW_;
  float* fb = feat + (size_t)b * 32 * 256;
  const int col = wave * 16 + m;       // channel
  const int rb  = (lane >> 4) * 8;     // D layout: lanes 16-31 hold rows r+8
#pragma unroll
  for (int r = 0; r < 8; ++r) {
    atomicAdd(&fb[(rb + r) * 256 + col],      acc0[r] * sc);
    atomicAdd(&fb[(16 + rb + r) * 256 + col], acc1[r] * sc);
  }
}

// ---------------- Stage 2: OUT = leaky(IN @ W + bias) via WMMA ------------
// IN: [256, Kin] row-major. W: [Kin, N] row-major. Grid: (16 Mtiles, ceil(N/256)).
#define TK   32
#define ISTR 34
#define WSTR 258

__global__ __launch_bounds__(512) void gemm_kernel(
    const float* __restrict__ IN, const float* __restrict__ W,
    const float* __restrict__ bias, float* __restrict__ OUT,
    int Kin, int N, int ldOut, float slope) {
  __shared__ float sIn[16 * ISTR];    //  2176 B
  __shared__ float sW[TK * WSTR];     // 33024 B
  const int tid  = threadIdx.x;
  const int lane = tid & 31;
  const int wave = tid >> 5;
  const int m    = lane & 15;
  const int hoff = (lane >> 4) << 1;
  const int mbase = blockIdx.x * 16;
  const int nbase = blockIdx.y * 256;
  const int nloc  = wave * 16 + m;

  v8f acc = {0, 0, 0, 0, 0, 0, 0, 0};

  for (int k0 = 0; k0 < Kin; k0 += TK) {
    __syncthreads();
    // IN tile 16x32 (one float per thread)
    sIn[wave * ISTR + lane] = IN[(size_t)(mbase + wave) * Kin + k0 + lane];
    // W tile 32x256 (guard n for the narrow classifier head)
#pragma unroll
    for (int i = 0; i < 16; ++i) {
      const int idx = tid + i * 512;
      const int k = idx >> 8;
      const int nl = idx & 255;
      const int n = nbase + nl;
      sW[k * WSTR + nl] = (n < N) ? W[(size_t)(k0 + k) * N + n] : 0.0f;
    }
    __syncthreads();
#pragma unroll
    for (int st = 0; st < TK / 4; ++st) {
      const int h = st * 4 + hoff;
      v2f a = *(const v2f*)&sIn[m * ISTR + h];
      v2f bb;
      bb.x = sW[h * WSTR + nloc];
      bb.y = sW[(h + 1) * WSTR + nloc];
      acc = wmma4(a, bb, acc);
    }
  }
  const int col = nbase + nloc;
  if (col < N) {
    const float bv = bias[col];
    const int rb = (lane >> 4) * 8;
#pragma unroll
    for (int r = 0; r < 8; ++r) {
      float v = acc[r] + bv;
      v = v > 0.0f ? v : slope * v;   // slope=1.0 -> identity (logits layer)
      OUT[(size_t)(mbase + rb + r) * ldOut + col] = v;
    }
  }
}

// ---------------- Stage 3: masked CE loss + accuracy ----------------------
__global__ __launch_bounds__(256) void loss_kernel(
    const float* __restrict__ logits /* [256 x 64], valid cols 0..37 */,
    const int* __restrict__ gt, const float* __restrict__ gt_src,
    const float* __restrict__ gt_tgt, const float* __restrict__ weights,
    float* __restrict__ out) {
  __shared__ float sCE[B_ * K_], sM[B_ * K_], sLS[B_ * K_], sMT[B_ * K_];
  __shared__ float sA[B_], sB[B_], sCt[B_];
  const int tid = threadIdx.x;
  if (tid < B_ * K_) {
    const int b = tid / K_, k = tid % K_;
    const float* lg = logits + (size_t)(b * 32 + k) * 64;
    float mx = -1e30f;
    for (int j = 0; j < 2 * K_; ++j) mx = fmaxf(mx, lg[j]);
    float se = 0.0f;
    for (int j = 0; j < 2 * K_; ++j) se += expf(lg[j] - mx);
    const float lse = logf(se) + mx;
    float ce = 0.0f, ls = 0.0f, bg = -1e30f, bl = -1e30f;
    int ag = 0, al = 0;
    for (int j = 0; j < 2 * K_; ++j) {
      const float l = (j < K_) ? gt_src[(size_t)(b * K_ + k) * K_ + j]
                               : gt_tgt[(size_t)(b * K_ + k) * K_ + (j - K_)];
      ce -= l * (lg[j] - lse);
      ls += l;
      if (lg[j] > bg) { bg = lg[j]; ag = j; }   // first-max, like argmax
      if (l > bl)     { bl = l;     al = j; }
    }
    const float msk = (gt[b * K_ + k] > 0) ? 1.0f : 0.0f;
    sCE[tid] = msk * weights[b * K_ + k] * ce;
    sM[tid]  = msk;
    sLS[tid] = msk * ls;
    sMT[tid] = msk * ((ag == al) ? 1.0f : 0.0f);
  }
  __syncthreads();
  if (tid < B_) {
    float sl = 0, n = 0, lsum = 0, mt = 0;
    for (int k = 0; k < K_; ++k) {
      const int i = tid * K_ + k;
      sl += sCE[i]; n += sM[i]; lsum += sLS[i]; mt += sMT[i];
    }
    sl /= fmaxf(n, 1.0f);
    const bool valid = lsum > 0.0f;
    sA[tid]  = valid ? sl : 0.0f;
    sB[tid]  = valid ? mt : 0.0f;
    sCt[tid] = valid ? lsum : 0.0f;
  }
  __syncthreads();
  if (tid == 0) {
    float loss = 0, acc = 0, cnt = 0;
    for (int b = 0; b < B_; ++b) { loss += sA[b]; acc += sB[b]; cnt += sCt[b]; }
    out[0] = loss / (float)B_;
    out[1] = acc / cnt * 100.0f;
  }
}

extern "C" void kernel_launch(void* const* d_in, const int* in_sizes, int n_in,
                              void* d_out, int out_size, void* d_ws, size_t ws_size,
                              hipStream_t stream) {
  const float* x   = (const float*)d_in[0];
  const float* cam = (const float*)d_in[1];
  const int*   gt  = (const int*)d_in[2];
  const float* gts = (const float*)d_in[3];
  const float* gtt = (const float*)d_in[4];
  const float* wts = (const float*)d_in[5];
  const float* W1  = (const float*)d_in[6];
  const float* b1  = (const float*)d_in[7];
  const float* W2  = (const float*)d_in[8];
  const float* b2  = (const float*)d_in[9];
  const float* Wc  = (const float*)d_in[10];
  const float* bc  = (const float*)d_in[11];
  float* out = (float*)d_out;

  float* ws     = (float*)d_ws;
  float* feat   = ws;                    // 256*256   (B*32 x 256, pad rows = 0)
  float* h1     = feat + 256 * 256;      // 256*2048
  float* h2     = h1   + 256 * 2048;     // 256*512
  float* logits = h2   + 256 * 512;      // 256*64

  zero_kernel<<<128, 256, 0, stream>>>(feat, 256 * 256);
  feat_kernel<<<dim3(64, B_), 512, 0, stream>>>(x, cam, feat);
  gemm_kernel<<<dim3(16, 8), 512, 0, stream>>>(feat, W1, b1, h1, 256, 2048, 2048, 0.2f);
  gemm_kernel<<<dim3(16, 2), 512, 0, stream>>>(h1, W2, b2, h2, 2048, 512, 512, 0.2f);
  gemm_kernel<<<dim3(16, 1), 512, 0, stream>>>(h2, Wc, bc, logits, 512, 38, 64, 1.0f);
  loss_kernel<<<1, 256, 0, stream>>>(logits, gt, gts, gtt, wts, out);
}